// CrossScaleAggregator_85452669321804
// MI455X (gfx1250) — compile-verified
//
#include <hip/hip_runtime.h>
#include <hip/hip_bf16.h>
#include <math.h>

#define BB 4
#define LL 2048
#define DD 256
#define NWIN 5
#define NTOT 6823      // 2048+2047+1023+1023+682
#define KTOP 10
#define MAXTILES 128
#define ALPHA_C 0.4f
#define BIGNEG -1e30f
#define IMAX 0x7fffffff

typedef float v2f __attribute__((ext_vector_type(2)));
typedef float v8f __attribute__((ext_vector_type(8)));

__device__ const int d_winoff[NWIN] = {0, 2048, 4095, 5118, 6141};

template<int K>
__device__ __forceinline__ void topk_insert(float (&tv)[K], int (&ti)[K], float v, int ix) {
#pragma unroll
  for (int k = 0; k < K; ++k) {
    if (v > tv[k] || (v == tv[k] && ix < ti[k])) {
#pragma unroll
      for (int m = K - 1; m > k; --m) { tv[m] = tv[m - 1]; ti[m] = ti[m - 1]; }
      tv[k] = v; ti[k] = ix;
      return;
    }
  }
}

// inv l2-norm per row of DD floats: inv = 1/max(||x||, 1e-12)
__global__ void row_inv_norm_kernel(const float* __restrict__ x, float* __restrict__ inv) {
  __shared__ float red[256];
  size_t row = blockIdx.x;
  float v = x[row * DD + threadIdx.x];
  red[threadIdx.x] = v * v;
  __syncthreads();
  for (int s = 128; s > 0; s >>= 1) {
    if (threadIdx.x < (unsigned)s) red[threadIdx.x] += red[threadIdx.x + s];
    __syncthreads();
  }
  if (threadIdx.x == 0) inv[row] = 1.0f / fmaxf(sqrtf(red[0]), 1e-12f);
}

// Greedy chain == top-4 of each cosine-sim row (excl. diagonal).
// f32 WMMA, quad accumulator chains (4 column tiles per wave iteration).
__global__ void chain_cooc_kernel(const float* __restrict__ tok,
                                  const float* __restrict__ tokInv,
                                  float* __restrict__ coocVec) {
  const int b = blockIdx.y;
  const int i0 = blockIdx.x * 16;
  const int tid = threadIdx.x;
  const int wave = tid >> 5;
  const int lane = tid & 31;
  const int half = lane >> 4;      // 0: rows M=0..7 / K=0,1   1: rows M=8..15 / K=2,3
  const int mrow = lane & 15;      // A row / B col / C col

  __shared__ float tileS[8][16][68];   // four 16x16 tiles per wave (17-stride cols)
  __shared__ float candV[8][16][4];
  __shared__ int   candI[8][16][4];
  __shared__ float pickV[16][4];
  __shared__ int   pickI[16][4];

  const float* tb  = tok + (size_t)b * LL * DD;
  const float* ivb = tokInv + (size_t)b * LL;

  const float* arow = tb + (size_t)(i0 + mrow) * DD + 2 * half;

  float invI[8];
#pragma unroll
  for (int r = 0; r < 8; ++r) invI[r] = ivb[i0 + r + 8 * half];

  float t4v[4] = {BIGNEG, BIGNEG, BIGNEG, BIGNEG};
  int   t4i[4] = {IMAX, IMAX, IMAX, IMAX};

  for (int jq = wave; jq < LL / 64; jq += 8) {          // uniform 4 iters per wave
    const int j0 = jq * 64;
    const float* brow0 = tb + (size_t)(j0 + mrow) * DD + 2 * half;
    const float* brow1 = tb + (size_t)(j0 + 16 + mrow) * DD + 2 * half;
    const float* brow2 = tb + (size_t)(j0 + 32 + mrow) * DD + 2 * half;
    const float* brow3 = tb + (size_t)(j0 + 48 + mrow) * DD + 2 * half;
    v8f acc0 = {}, acc1 = {}, acc2 = {}, acc3 = {};
#pragma unroll 16
    for (int kk = 0; kk < DD; kk += 4) {
      v2f av  = *(const v2f*)(arow + kk);
      v2f bv0 = *(const v2f*)(brow0 + kk);
      v2f bv1 = *(const v2f*)(brow1 + kk);
      v2f bv2 = *(const v2f*)(brow2 + kk);
      v2f bv3 = *(const v2f*)(brow3 + kk);
      acc0 = __builtin_amdgcn_wmma_f32_16x16x4_f32(false, av, false, bv0,
                                                   (short)0, acc0, false, false);
      acc1 = __builtin_amdgcn_wmma_f32_16x16x4_f32(false, av, false, bv1,
                                                   (short)0, acc1, false, false);
      acc2 = __builtin_amdgcn_wmma_f32_16x16x4_f32(false, av, false, bv2,
                                                   (short)0, acc2, false, false);
      acc3 = __builtin_amdgcn_wmma_f32_16x16x4_f32(false, av, false, bv3,
                                                   (short)0, acc3, false, false);
    }
    const float invJ0 = ivb[j0 + mrow];
    const float invJ1 = ivb[j0 + 16 + mrow];
    const float invJ2 = ivb[j0 + 32 + mrow];
    const float invJ3 = ivb[j0 + 48 + mrow];
#pragma unroll
    for (int r = 0; r < 8; ++r) {
      const int m = r + 8 * half;
      tileS[wave][m][mrow]      = acc0[r] * invI[r] * invJ0;
      tileS[wave][m][17 + mrow] = acc1[r] * invI[r] * invJ1;
      tileS[wave][m][34 + mrow] = acc2[r] * invI[r] * invJ2;
      tileS[wave][m][51 + mrow] = acc3[r] * invI[r] * invJ3;
    }
    __syncthreads();
    if (lane < 16) {
      const int i = i0 + lane;
      for (int c = 0; c < 64; ++c) {
        const int j = j0 + c;
        if (j == i) continue;                            // diagonal excluded (eye/-9)
        const float v = tileS[wave][lane][(c & 15) + 17 * (c >> 4)];
        topk_insert<4>(t4v, t4i, v, j);
      }
    }
    __syncthreads();
  }

  if (lane < 16) {
#pragma unroll
    for (int k = 0; k < 4; ++k) { candV[wave][lane][k] = t4v[k]; candI[wave][lane][k] = t4i[k]; }
  }
  __syncthreads();
  if (tid < 16) {
    float fv[4] = {BIGNEG, BIGNEG, BIGNEG, BIGNEG};
    int   fi[4] = {IMAX, IMAX, IMAX, IMAX};
    for (int wv = 0; wv < 8; ++wv)
      for (int k = 0; k < 4; ++k)
        topk_insert<4>(fv, fi, candV[wv][tid][k], candI[wv][tid][k]);
#pragma unroll
    for (int k = 0; k < 4; ++k) { pickV[tid][k] = fv[k]; pickI[tid][k] = fi[k]; }
  }
  __syncthreads();

  // cooc row = (self + 4 picks) / 5 ; pick falls back to self when val < ALPHA
  const int r  = tid >> 4;
  const int e0 = (tid & 15) * 16;
  const int i  = i0 + r;
  float av[16];
  const float* self = tb + (size_t)i * DD + e0;
#pragma unroll
  for (int e = 0; e < 16; ++e) av[e] = self[e];
  for (int t = 0; t < 4; ++t) {
    const int idx = (pickV[r][t] >= ALPHA_C) ? pickI[r][t] : i;
    const float* pr = tb + (size_t)idx * DD + e0;
#pragma unroll
    for (int e = 0; e < 16; ++e) av[e] += pr[e];
  }
  float* orow = coocVec + ((size_t)b * LL + i) * DD + e0;
#pragma unroll
  for (int e = 0; e < 16; ++e) orow[e] = av[e] * 0.2f;
}

__global__ void pos_mean_kernel(const float* __restrict__ tok, float* __restrict__ posVec,
                                int w, int s, int off) {
  const int p = blockIdx.x, b = blockIdx.y, e = threadIdx.x;
  const float* tb = tok + (size_t)b * LL * DD;
  float acc = 0.f;
  const int base = p * s;
  for (int t = 0; t < w; ++t) acc += tb[(size_t)(base + t) * DD + e];
  posVec[((size_t)b * NTOT + off + p) * DD + e] = acc / (float)w;
}

// iou tile GEMM via f32 WMMA (quad accumulator chains) + per-lane top-10
__global__ void iou_topk_kernel(const float* __restrict__ posVec,
                                const float* __restrict__ posInv,
                                const float* __restrict__ coocVec,
                                const float* __restrict__ coocInv,
                                float* __restrict__ partV, int* __restrict__ partI,
                                int n_pos, int off) {
  const int b = blockIdx.y;
  const int p0 = blockIdx.x * 16;
  const int tid = threadIdx.x;
  const int wave = tid >> 5, lane = tid & 31;
  const int half = lane >> 4, mrow = lane & 15;

  const float* pbase = posVec + ((size_t)b * NTOT + off) * DD;
  const float* pinvb = posInv + (size_t)b * NTOT + off;
  const float* cbase = coocVec + (size_t)b * LL * DD;
  const float* cinvb = coocInv + (size_t)b * LL;

  const int prow = p0 + mrow;
  const int prowC = (prow < n_pos) ? prow : (n_pos - 1);   // clamp pad rows
  const float* arow = pbase + (size_t)prowC * DD + 2 * half;

  float invP[8];
#pragma unroll
  for (int r = 0; r < 8; ++r) {
    const int pi = p0 + r + 8 * half;
    invP[r] = (pi < n_pos) ? pinvb[pi] : 0.f;
  }

  float tv[KTOP]; int tix[KTOP];
#pragma unroll
  for (int k = 0; k < KTOP; ++k) { tv[k] = BIGNEG; tix[k] = IMAX; }

  for (int jq = wave; jq < LL / 64; jq += 8) {
    const int j0 = jq * 64;
    const float* brow0 = cbase + (size_t)(j0 + mrow) * DD + 2 * half;
    const float* brow1 = cbase + (size_t)(j0 + 16 + mrow) * DD + 2 * half;
    const float* brow2 = cbase + (size_t)(j0 + 32 + mrow) * DD + 2 * half;
    const float* brow3 = cbase + (size_t)(j0 + 48 + mrow) * DD + 2 * half;
    v8f acc0 = {}, acc1 = {}, acc2 = {}, acc3 = {};
#pragma unroll 16
    for (int kk = 0; kk < DD; kk += 4) {
      v2f av  = *(const v2f*)(arow + kk);
      v2f bv0 = *(const v2f*)(brow0 + kk);
      v2f bv1 = *(const v2f*)(brow1 + kk);
      v2f bv2 = *(const v2f*)(brow2 + kk);
      v2f bv3 = *(const v2f*)(brow3 + kk);
      acc0 = __builtin_amdgcn_wmma_f32_16x16x4_f32(false, av, false, bv0,
                                                   (short)0, acc0, false, false);
      acc1 = __builtin_amdgcn_wmma_f32_16x16x4_f32(false, av, false, bv1,
                                                   (short)0, acc1, false, false);
      acc2 = __builtin_amdgcn_wmma_f32_16x16x4_f32(false, av, false, bv2,
                                                   (short)0, acc2, false, false);
      acc3 = __builtin_amdgcn_wmma_f32_16x16x4_f32(false, av, false, bv3,
                                                   (short)0, acc3, false, false);
    }
    const float invJ0 = cinvb[j0 + mrow];
    const float invJ1 = cinvb[j0 + 16 + mrow];
    const float invJ2 = cinvb[j0 + 32 + mrow];
    const float invJ3 = cinvb[j0 + 48 + mrow];
    const int ci0 = j0 + mrow;
#pragma unroll
    for (int r = 0; r < 8; ++r) {
      const int pi = p0 + r + 8 * half;
      if (pi >= n_pos) continue;
      const int fb = pi * LL + ci0;
      topk_insert<KTOP>(tv, tix, acc0[r] * invP[r] * invJ0, fb);
      topk_insert<KTOP>(tv, tix, acc1[r] * invP[r] * invJ1, fb + 16);
      topk_insert<KTOP>(tv, tix, acc2[r] * invP[r] * invJ2, fb + 32);
      topk_insert<KTOP>(tv, tix, acc3[r] * invP[r] * invJ3, fb + 48);
    }
  }

  __shared__ float cV[256][KTOP];
  __shared__ int   cI[256][KTOP];
#pragma unroll
  for (int k = 0; k < KTOP; ++k) { cV[tid][k] = tv[k]; cI[tid][k] = tix[k]; }
  __syncthreads();
  if (tid == 0) {
    float fv[KTOP]; int fi[KTOP];
#pragma unroll
    for (int k = 0; k < KTOP; ++k) { fv[k] = BIGNEG; fi[k] = IMAX; }
    for (int t = 0; t < 256; ++t)
      for (int k = 0; k < KTOP; ++k) {
        if (cV[t][k] <= -1e29f) break;
        topk_insert<KTOP>(fv, fi, cV[t][k], cI[t][k]);
      }
    const size_t base = ((size_t)b * MAXTILES + blockIdx.x) * KTOP;
#pragma unroll
    for (int k = 0; k < KTOP; ++k) { partV[base + k] = fv[k]; partI[base + k] = fi[k]; }
  }
}

__global__ void merge_topk_kernel(const float* __restrict__ partV, const int* __restrict__ partI,
                                  int n_tiles, int* __restrict__ finalIdx, int win) {
  const int b = blockIdx.x;
  if (threadIdx.x != 0) return;
  float fv[KTOP]; int fi[KTOP];
#pragma unroll
  for (int k = 0; k < KTOP; ++k) { fv[k] = BIGNEG; fi[k] = IMAX; }
  const size_t base = (size_t)b * MAXTILES * KTOP;
  for (int t = 0; t < n_tiles * KTOP; ++t) {
    const float v = partV[base + t];
    if (v <= -1e29f) continue;
    topk_insert<KTOP>(fv, fi, v, partI[base + t]);
  }
#pragma unroll
  for (int k = 0; k < KTOP; ++k) finalIdx[((size_t)b * NWIN + win) * KTOP + k] = fi[k];
}

// u = [p ; c] @ W^T + b, then row l2norm
__global__ void unit_kernel(const float* __restrict__ posVec, const float* __restrict__ coocVec,
                            const int* __restrict__ finalIdx,
                            const float* __restrict__ W, const float* __restrict__ bias,
                            float* __restrict__ out) {
  const int b = blockIdx.y;
  const int u = blockIdx.x;                 // 0..49
  const int win = u / KTOP, k = u % KTOP;
  const int e = threadIdx.x;
  const int flat = finalIdx[((size_t)b * NWIN + win) * KTOP + k];
  const int pi = flat / LL, ci = flat % LL;
  const float* p = posVec + ((size_t)b * NTOT + d_winoff[win] + pi) * DD;
  const float* c = coocVec + ((size_t)b * LL + ci) * DD;
  const float* wr = W + (size_t)e * (2 * DD);
  float acc = bias[e];
  for (int j = 0; j < DD; ++j) acc += p[j] * wr[j];
  for (int j = 0; j < DD; ++j) acc += c[j] * wr[DD + j];
  __shared__ float red[256];
  __shared__ float invn;
  red[e] = acc * acc;
  __syncthreads();
  for (int s = 128; s > 0; s >>= 1) {
    if (e < s) red[e] += red[e + s];
    __syncthreads();
  }
  if (e == 0) invn = 1.0f / fmaxf(sqrtf(red[0]), 1e-12f);
  __syncthreads();
  out[((size_t)b * (NWIN * KTOP) + u) * DD + e] = acc * invn;
}

extern "C" void kernel_launch(void* const* d_in, const int* in_sizes, int n_in,
                              void* d_out, int out_size, void* d_ws, size_t ws_size,
                              hipStream_t stream) {
  const float* tok  = (const float*)d_in[0];
  const float* W    = (const float*)d_in[1];
  const float* bias = (const float*)d_in[2];
  float* out = (float*)d_out;

  float* ws = (float*)d_ws;
  size_t o = 0;
  float* tokInv   = ws + o; o += (size_t)BB * LL;
  float* coocVec  = ws + o; o += (size_t)BB * LL * DD;
  float* coocInv  = ws + o; o += (size_t)BB * LL;
  float* posVec   = ws + o; o += (size_t)BB * NTOT * DD;
  float* posInv   = ws + o; o += (size_t)BB * NTOT;
  float* partV    = ws + o; o += (size_t)BB * MAXTILES * KTOP;
  int*   partI    = (int*)(ws + o); o += (size_t)BB * MAXTILES * KTOP;
  int*   finalIdx = (int*)(ws + o); o += (size_t)BB * NWIN * KTOP;

  static const int wlist[NWIN]   = {1, 2, 3, 4, 5};
  static const int slist[NWIN]   = {1, 1, 2, 2, 3};
  static const int nplist[NWIN]  = {2048, 2047, 1023, 1023, 682};
  static const int offlist[NWIN] = {0, 2048, 4095, 5118, 6141};

  // 1) token inv-norms
  row_inv_norm_kernel<<<BB * LL, 256, 0, stream>>>(tok, tokInv);
  // 2) greedy chain via streamed f32-WMMA sim tiles -> cooc_vec
  chain_cooc_kernel<<<dim3(LL / 16, BB), 256, 0, stream>>>(tok, tokInv, coocVec);
  // 3) cooc inv-norms
  row_inv_norm_kernel<<<BB * LL, 256, 0, stream>>>(coocVec, coocInv);
  // 4) window means
  for (int w = 0; w < NWIN; ++w)
    pos_mean_kernel<<<dim3(nplist[w], BB), 256, 0, stream>>>(tok, posVec, wlist[w], slist[w], offlist[w]);
  // 5) pos inv-norms
  row_inv_norm_kernel<<<BB * NTOT, 256, 0, stream>>>(posVec, posInv);
  // 6) per-window iou GEMM + top-10
  for (int w = 0; w < NWIN; ++w) {
    const int tiles = (nplist[w] + 15) / 16;
    iou_topk_kernel<<<dim3(tiles, BB), 256, 0, stream>>>(posVec, posInv, coocVec, coocInv,
                                                         partV, partI, nplist[w], offlist[w]);
    merge_topk_kernel<<<BB, 32, 0, stream>>>(partV, partI, tiles, finalIdx, w);
  }
  // 7) linear + l2norm -> out (B, 50, 256)
  unit_kernel<<<dim3(NWIN * KTOP, BB), 256, 0, stream>>>(posVec, coocVec, finalIdx, W, bias, out);
}